// MeshEstimator_18820546691728
// MI455X (gfx1250) — compile-verified
//
#include <hip/hip_runtime.h>
#include <math.h>

#define PI_F 3.14159265358979323846f

typedef __attribute__((ext_vector_type(2))) float v2f;
typedef __attribute__((ext_vector_type(8))) float v8f;

#define BATCH 256
#define RVERT 6890
#define NCOLS (RVERT * 3)   /* 20670 */
#define NPAD  20672
#define KTOT  440
#define XDIM  88

/* ---- d_out element offsets (return-order concat) ---- */
#define OUT_BETAS  0
#define OUT_POSE   (OUT_BETAS + BATCH * 10)
#define OUT_RSHIFT (OUT_POSE + BATCH * 72)
#define OUT_RANG   (OUT_RSHIFT + BATCH * 3)
#define OUT_VERTS  (OUT_RANG + BATCH * 6)
#define OUT_VRED   (OUT_VERTS + BATCH * NCOLS)
#define OUT_NEWJ   (OUT_VRED + BATCH * 30)
#define OUT_VOFF   (OUT_NEWJ + BATCH * 72)

/* ---- workspace element offsets ---- */
#define WS_F   0
#define WS_RS  (WS_F + BATCH * KTOT)            /* 24*9 per batch */
#define WS_J   (WS_RS + BATCH * 216)            /* 24*3 per batch */
#define WS_A   (WS_J + BATCH * 72)              /* 24*12 per batch */
#define WS_VS  (WS_A + BATCH * 288)             /* v_shaped BATCH x NPAD */
#define WS_VP  (WS_VS + (size_t)BATCH * NPAD)   /* v_posed  BATCH x NPAD */
#define WS_WP  (WS_VP + (size_t)BATCH * NPAD)   /* packed W, col-major NPAD x KTOT */

__constant__ int PARENTS_D[24] = {0,0,0,0,1,2,3,4,5,6,7,8,9,9,9,12,13,14,16,17,18,19,20,21};
__constant__ int VLIST_D[10]   = {1325,336,1032,4515,1374,4848,1739,5209,1960,5423};
__constant__ int SYNTH_D[10]   = {3,15,4,5,7,8,18,19,20,21};

__constant__ float BLO[72] = {
  -0.5933865286111969f, -2.0f*PI_F, -1.215762200416361f,
  -1.5793940868065197f, -0.5881754611f, -0.5323249722f,
  -1.5793940868065197f, -0.5689768556f, -0.6736965222f,
  -PI_F/3.0f, -PI_F/36.0f, -PI_F/36.0f,
  -0.02268926111f, -0.01f, -0.01f,
  -0.02268926111f, -0.01f, -0.01f,
  -PI_F/3.0f, -PI_F/36.0f, -PI_F/36.0f,
  -PI_F/6.0f, -PI_F/6.0f, -PI_F/6.0f, -PI_F/6.0f, -PI_F/6.0f, -PI_F/6.0f,
  -PI_F/3.0f, -PI_F/36.0f, -PI_F/36.0f,
  -0.01f, -0.01f, -0.01f, -0.01f, -0.01f, -0.01f,
  -PI_F/3.0f, -PI_F/36.0f, -PI_F/36.0f,
  -1.551596394f/3.0f, -2.455676183f/3.0f, -1.570795f/3.0f,
  -1.551596394f/3.0f, -0.7627082389f/3.0f, -2.188641033f/3.0f,
  -PI_F/3.0f, -PI_F/36.0f, -PI_F/36.0f,
  -1.551596394f*2.0f/3.0f, -2.455676183f*2.0f/3.0f, -1.570795f*2.0f/3.0f,
  -1.551596394f*2.0f/3.0f, -0.7627082389f*2.0f/3.0f, -2.188641033f*2.0f/3.0f,
  -0.01f, -2.570867817f, -0.01f, -0.01f, -0.04799651389f, -0.01f,
  -PI_F/6.0f, -PI_F/6.0f, -PI_F/6.0f, -PI_F/6.0f, -PI_F/6.0f, -PI_F/6.0f,
  -0.01f, -0.01f, -0.01f, -0.01f, -0.01f, -0.01f
};
__constant__ float BHI[72] = {
  0.5933865286111969f, 2.0f*PI_F, 1.215762200416361f,
  0.3097956806f, 0.5689768556f, 0.6736965222f,
  0.3097956806f, 0.5881754611f, 0.5323249722f,
  PI_F/3.0f, PI_F/36.0f, PI_F/36.0f,
  2.441713561f, 0.01f, 0.01f,
  2.441713561f, 0.01f, 0.01f,
  PI_F/3.0f, PI_F/36.0f, PI_F/36.0f,
  PI_F/6.0f, PI_F/6.0f, PI_F/6.0f, PI_F/6.0f, PI_F/6.0f, PI_F/6.0f,
  PI_F/3.0f, PI_F/36.0f, PI_F/36.0f,
  0.01f, 0.01f, 0.01f, 0.01f, 0.01f, 0.01f,
  PI_F/3.0f, PI_F/36.0f, PI_F/36.0f,
  2.206094311f/3.0f, 0.7627082389f/3.0f, 2.188641033f/3.0f,
  2.206094311f/3.0f, 2.455676183f/3.0f, 1.570795f/3.0f,
  PI_F/3.0f, PI_F/36.0f, PI_F/36.0f,
  2.206094311f*2.0f/3.0f, 0.7627082389f*2.0f/3.0f, 2.188641033f*2.0f/3.0f,
  2.206094311f*2.0f/3.0f, 2.455676183f*2.0f/3.0f, 1.570795f*2.0f/3.0f,
  0.01f, 0.04799651389f, 0.01f, 0.01f, 2.570867817f, 0.01f,
  PI_F/6.0f, PI_F/6.0f, PI_F/6.0f, PI_F/6.0f, PI_F/6.0f, PI_F/6.0f,
  0.01f, 0.01f, 0.01f, 0.01f, 0.01f, 0.01f
};

/* W(k, col) gathered from original input layouts */
__device__ __forceinline__ float loadW(int k, int col, const float* __restrict__ vt,
                                       const float* __restrict__ sd, const float* __restrict__ pd) {
  if (k < 2)  return vt[(size_t)k * NCOLS + col];           /* v_template rows g0,g1 */
  if (k < 22) return sd[(size_t)(k - 2) * NCOLS + col];     /* shapedirs: (k-2) valid for both genders */
  if (k < 24) return 0.0f;                                  /* pad */
  if (k < 232) { int p = k - 24;  return (p < 207) ? pd[(size_t)col * 207 + p] : 0.0f; }
  { int p = k - 232; return (p < 207) ? pd[(size_t)NCOLS * 207 + (size_t)col * 207 + p] : 0.0f; }
}

/* ============ K0: pack W column-major (NPAD x KTOT) -> branchless GEMM B loads ============ */
__global__ void __launch_bounds__(64) k0_pack(const float* __restrict__ vt,
                                              const float* __restrict__ sd,
                                              const float* __restrict__ pd,
                                              float* __restrict__ wp) {
  int col = blockIdx.x;                      /* 0..NPAD-1 */
  int colc = col < NCOLS ? col : NCOLS - 1;  /* padded cols: value irrelevant (results discarded) */
  float* dst = wp + (size_t)col * KTOT;
  for (int k = threadIdx.x; k < KTOT; k += 64) dst[k] = loadW(k, colc, vt, sd, pd);
}

/* ============ K1: per-batch scalar prep ============ */
__global__ void k1_prep(const float* __restrict__ x, const float* __restrict__ gender,
                        float* __restrict__ out, float* __restrict__ ws) {
  int b = blockIdx.x * blockDim.x + threadIdx.x;
  if (b >= BATCH) return;
  const float* xr = x + b * XDIM;
  float g0 = gender[b * 2 + 0], g1 = gender[b * 2 + 1];

  float betas[10];
  for (int k = 0; k < 10; k++) {
    betas[k] = tanhf(xr[k] / 3.0f) * 3.0f;
    out[OUT_BETAS + b * 10 + k] = betas[k];
  }
  float rs[3] = { xr[10] + 0.314f, xr[11] + 0.914f, xr[12] + 0.1f };
  for (int d = 0; d < 3; d++) out[OUT_RSHIFT + b * 3 + d] = rs[d];
  for (int k = 0; k < 6; k++) out[OUT_RANG + b * 6 + k] = xr[13 + k];

  float pose[72];
  pose[0] = atan2f(xr[16], xr[13]);
  pose[1] = atan2f(xr[17], xr[14]);
  pose[2] = atan2f(xr[18], xr[15]);
  for (int k = 3; k < 72; k++) pose[k] = xr[16 + k]; /* x[19 + (k-3)] */
  for (int k = 0; k < 72; k++) {
    float lo = BLO[k], hi = BHI[k];
    float m = lo + hi;          /* bounds = 2*b -> mean = lo+hi   */
    float s = 1.0f / (hi - lo); /* scale = 2/|2lo-2hi| = 1/(hi-lo) */
    float p = tanhf((pose[k] - m) * s) / s + m;
    pose[k] = p;
    out[OUT_POSE + b * 72 + k] = p;
  }

  float* Rs = ws + WS_RS + (size_t)b * 216;
  float* Fr = ws + WS_F + (size_t)b * KTOT;
  Fr[0] = g0; Fr[1] = g1;
  for (int k = 0; k < 10; k++) { Fr[2 + k] = g0 * betas[k]; Fr[12 + k] = g1 * betas[k]; }
  Fr[22] = 0.0f; Fr[23] = 0.0f; Fr[231] = 0.0f; Fr[439] = 0.0f;

  for (int i = 0; i < 24; i++) {
    float tx = pose[3 * i], ty = pose[3 * i + 1], tz = pose[3 * i + 2];
    float ax = tx + 1e-8f, ay = ty + 1e-8f, az = tz + 1e-8f;
    float ang = sqrtf(ax * ax + ay * ay + az * az);
    float hlf = 0.5f * ang;
    float sh = sinf(hlf), ch = cosf(hlf);
    float inva = 1.0f / ang;
    float qw = ch, qx = tx * inva * sh, qy = ty * inva * sh, qz = tz * inva * sh;
    float qn = 1.0f / sqrtf(qw * qw + qx * qx + qy * qy + qz * qz);
    qw *= qn; qx *= qn; qy *= qn; qz *= qn;
    float Rm[9];
    Rm[0] = 1.0f - 2.0f * (qy * qy + qz * qz); Rm[1] = 2.0f * (qx * qy - qw * qz); Rm[2] = 2.0f * (qx * qz + qw * qy);
    Rm[3] = 2.0f * (qx * qy + qw * qz); Rm[4] = 1.0f - 2.0f * (qx * qx + qz * qz); Rm[5] = 2.0f * (qy * qz - qw * qx);
    Rm[6] = 2.0f * (qx * qz - qw * qy); Rm[7] = 2.0f * (qy * qz + qw * qx); Rm[8] = 1.0f - 2.0f * (qx * qx + qy * qy);
    for (int e = 0; e < 9; e++) Rs[i * 9 + e] = Rm[e];
    if (i >= 1) {
      int base = (i - 1) * 9;
      for (int e = 0; e < 9; e++) {
        float pf = Rm[e] - ((e == 0 || e == 4 || e == 8) ? 1.0f : 0.0f);
        Fr[24 + base + e]  = g0 * pf;
        Fr[232 + base + e] = g1 * pf;
      }
    }
  }
}

/* ============ K2: branchless WMMA GEMM v_posed = F(256x440) @ W(440x20670) ============ */
__device__ __forceinline__ void store_rows(float* __restrict__ dst, v8f a, int mrow0, int col) {
#pragma unroll
  for (int v = 0; v < 8; v++) dst[(size_t)(mrow0 + v) * NPAD + col] = a[v];
}

__global__ void __launch_bounds__(32) k2_gemm(const float* __restrict__ F,
                                              const float* __restrict__ wp,
                                              float* __restrict__ vs,
                                              float* __restrict__ vp) {
  int lane = threadIdx.x;
  int half = lane >> 4;       /* 0: K pair {0,1} | 1: K pair {2,3} */
  int l16  = lane & 15;
  int nt = blockIdx.x * 16;   /* column tile */
  int mb = blockIdx.y * 64;   /* 4 row tiles of 16 */
  int col = nt + l16;

  const float* Wc = wp + (size_t)col * KTOT;   /* 8B-aligned: 440*4 = 1760 bytes */
  const float* Fp[4];
#pragma unroll
  for (int mt = 0; mt < 4; mt++) Fp[mt] = F + (size_t)(mb + mt * 16 + l16) * KTOT;

  v8f acc[4] = {};

#pragma unroll
  for (int k = 0; k < KTOT; k += 4) {
    int kb0 = k + half * 2;                     /* even -> 8B-aligned v2f loads */
    v2f bf = *(const v2f*)(Wc + kb0);
#pragma unroll
    for (int mt = 0; mt < 4; mt++) {
      v2f af = *(const v2f*)(Fp[mt] + kb0);
      acc[mt] = __builtin_amdgcn_wmma_f32_16x16x4_f32(false, af, false, bf,
                                                      (short)0, acc[mt], false, false);
    }
    if (k == 20) { /* first 24 K-values done -> snapshot = v_shaped */
      if (col < NCOLS) {
#pragma unroll
        for (int mt = 0; mt < 4; mt++) store_rows(vs, acc[mt], mb + mt * 16 + half * 8, col);
      }
    }
  }
  if (col < NCOLS) {
#pragma unroll
    for (int mt = 0; mt < 4; mt++) store_rows(vp, acc[mt], mb + mt * 16 + half * 8, col);
  }
}

/* ============ K3: J[b,j,d] = sum_r mix(J_regressor) * v_shaped ============ */
__global__ void k3_jreg(const float* __restrict__ gender, const float* __restrict__ jr,
                        const float* __restrict__ vs, float* __restrict__ Jout) {
  int idx = blockIdx.x * blockDim.x + threadIdx.x;
  if (idx >= BATCH * 72) return;
  int b = idx / 72, jd = idx % 72, j = jd / 3, d = jd % 3;
  float g0 = gender[b * 2 + 0], g1 = gender[b * 2 + 1];
  const float* vsb = vs + (size_t)b * NPAD;
  float s = 0.0f;
  for (int r = 0; r < RVERT; r++) {
    float wr = g0 * jr[(size_t)r * 24 + j] + g1 * jr[(size_t)RVERT * 24 + (size_t)r * 24 + j];
    s += wr * vsb[r * 3 + d];
  }
  Jout[idx] = s;
}

/* ============ K4: kinematic chain -> A matrices + new_J ============ */
__global__ void k4_rigid(float* __restrict__ ws, float* __restrict__ out) {
  int b = blockIdx.x * blockDim.x + threadIdx.x;
  if (b >= BATCH) return;
  const float* Rs = ws + WS_RS + (size_t)b * 216;
  const float* J  = ws + WS_J + (size_t)b * 72;
  float* A = ws + WS_A + (size_t)b * 288;
  const float* rsh = out + OUT_RSHIFT + b * 3;

  float G[24][12];
  for (int rr = 0; rr < 3; rr++) {
    for (int cc = 0; cc < 3; cc++) G[0][rr * 4 + cc] = Rs[rr * 3 + cc];
    G[0][rr * 4 + 3] = J[rr];
  }
  for (int i = 1; i < 24; i++) {
    int p = PARENTS_D[i];
    float t[3];
    for (int d = 0; d < 3; d++) t[d] = J[i * 3 + d] - J[p * 3 + d];
    const float* Ri = Rs + i * 9;
    for (int rr = 0; rr < 3; rr++) {
      float a0 = G[p][rr * 4 + 0], a1 = G[p][rr * 4 + 1], a2 = G[p][rr * 4 + 2];
      for (int cc = 0; cc < 3; cc++)
        G[i][rr * 4 + cc] = a0 * Ri[cc] + a1 * Ri[3 + cc] + a2 * Ri[6 + cc];
      G[i][rr * 4 + 3] = a0 * t[0] + a1 * t[1] + a2 * t[2] + G[p][rr * 4 + 3];
    }
  }
  for (int i = 0; i < 24; i++) {
    for (int rr = 0; rr < 3; rr++) {
      float gj = G[i][rr * 4 + 0] * J[i * 3 + 0] + G[i][rr * 4 + 1] * J[i * 3 + 1] +
                 G[i][rr * 4 + 2] * J[i * 3 + 2];
      A[i * 12 + rr * 4 + 0] = G[i][rr * 4 + 0];
      A[i * 12 + rr * 4 + 1] = G[i][rr * 4 + 1];
      A[i * 12 + rr * 4 + 2] = G[i][rr * 4 + 2];
      A[i * 12 + rr * 4 + 3] = G[i][rr * 4 + 3] - gj;
      out[OUT_NEWJ + b * 72 + i * 3 + rr] = G[i][rr * 4 + 3] - J[rr] + rsh[rr];
    }
  }
}

/* ============ K5: linear blend skinning ============ */
__global__ void __launch_bounds__(256) k5_skin(const float* __restrict__ gender,
                                               const float* __restrict__ wts,
                                               const float* __restrict__ ws,
                                               float* __restrict__ out) {
  int b = blockIdx.x;
  __shared__ float Alds[288];
  const float* A = ws + WS_A + (size_t)b * 288;
  for (int i = threadIdx.x; i < 288; i += 256) Alds[i] = A[i];
  __syncthreads();
  float g0 = gender[b * 2 + 0], g1 = gender[b * 2 + 1];
  float j0x = ws[WS_J + b * 72 + 0], j0y = ws[WS_J + b * 72 + 1], j0z = ws[WS_J + b * 72 + 2];
  float r0 = out[OUT_RSHIFT + b * 3 + 0], r1 = out[OUT_RSHIFT + b * 3 + 1], r2 = out[OUT_RSHIFT + b * 3 + 2];
  const float* vp = ws + WS_VP + (size_t)b * NPAD;

  for (int r = threadIdx.x; r < RVERT; r += 256) {
    float T[12];
#pragma unroll
    for (int e = 0; e < 12; e++) T[e] = 0.0f;
    const float* w0 = wts + (size_t)r * 24;
    const float* w1 = wts + (size_t)RVERT * 24 + (size_t)r * 24;
    for (int j = 0; j < 24; j++) {
      float wj = g0 * w0[j] + g1 * w1[j];
      const float* Aj = Alds + j * 12;
#pragma unroll
      for (int e = 0; e < 12; e++) T[e] += wj * Aj[e];
    }
    float vx = vp[r * 3 + 0], vy = vp[r * 3 + 1], vz = vp[r * 3 + 2];
    float ox = T[0] * vx + T[1] * vy + T[2]  * vz + T[3]  - j0x + r0;
    float oy = T[4] * vx + T[5] * vy + T[6]  * vz + T[7]  - j0y + r1;
    float oz = T[8] * vx + T[9] * vy + T[10] * vz + T[11] - j0z + r2;
    size_t o = (size_t)OUT_VERTS + ((size_t)b * RVERT + r) * 3;
    out[o + 0] = ox; out[o + 1] = oy; out[o + 2] = oz;
  }
}

/* ============ K6: verts_red + verts_offset ============ */
__global__ void k6_red(float* __restrict__ out) {
  int b = blockIdx.x;
  int i = threadIdx.x;
  if (i < 30) {
    int v = i / 3, d = i % 3;
    int r = VLIST_D[v];
    float val = out[(size_t)OUT_VERTS + ((size_t)b * RVERT + r) * 3 + d];
    out[OUT_VRED + b * 30 + i] = val;
    int jn = SYNTH_D[v];
    out[OUT_VOFF + b * 30 + i] = val - out[OUT_NEWJ + b * 72 + jn * 3 + d];
  }
}

extern "C" void kernel_launch(void* const* d_in, const int* in_sizes, int n_in,
                              void* d_out, int out_size, void* d_ws, size_t ws_size,
                              hipStream_t stream) {
  (void)in_sizes; (void)n_in; (void)out_size; (void)ws_size;
  const float* x  = (const float*)d_in[0];
  const float* g  = (const float*)d_in[1];
  const float* vt = (const float*)d_in[2];
  const float* sd = (const float*)d_in[3];
  const float* jr = (const float*)d_in[4];
  const float* pd = (const float*)d_in[5];
  const float* wt = (const float*)d_in[6];
  float* out = (float*)d_out;
  float* ws  = (float*)d_ws;

  k0_pack<<<dim3(NPAD), dim3(64), 0, stream>>>(vt, sd, pd, ws + WS_WP);
  k1_prep<<<dim3(8), dim3(32), 0, stream>>>(x, g, out, ws);
  k2_gemm<<<dim3(NPAD / 16, 4), dim3(32), 0, stream>>>(ws + WS_F, ws + WS_WP,
                                                       ws + WS_VS, ws + WS_VP);
  k3_jreg<<<dim3((BATCH * 72 + 127) / 128), dim3(128), 0, stream>>>(g, jr, ws + WS_VS, ws + WS_J);
  k4_rigid<<<dim3(8), dim3(32), 0, stream>>>(ws, out);
  k5_skin<<<dim3(BATCH), dim3(256), 0, stream>>>(g, wt, ws, out);
  k6_red<<<dim3(BATCH), dim3(32), 0, stream>>>(out);
}